// PhaseInvariantPearsonLoss_68856915689812
// MI455X (gfx1250) — compile-verified
//
#include <hip/hip_runtime.h>
#include <hip/hip_bf16.h>

typedef __attribute__((ext_vector_type(2))) float v2f;
typedef __attribute__((ext_vector_type(8))) float v8f;

#define TLEN 2048
#define MAXSHIFT 15
#define EPSV 1e-8f

// One wave (32 threads) per row. Computes best shifted Pearson per row.
__global__ __launch_bounds__(32) void pearson_rows_kernel(
    const float* __restrict__ preds,
    const float* __restrict__ targets,
    float* __restrict__ best_out) {
  __shared__ __align__(16) float tcExt[TLEN + 32];   // targets with +/-16 circular halo
  __shared__ __align__(16) float pcL[TLEN];          // preds row
  __shared__ __align__(16) float G0s[256];           // lag tile, A offset 0
  __shared__ __align__(16) float GPs[256];           // lag tile, A offset +16
  __shared__ __align__(16) float GMs[256];           // lag tile, A offset -16

  const int row  = blockIdx.x;
  const int lane = threadIdx.x;
  const float* prow = preds   + (size_t)row * TLEN;
  const float* trow = targets + (size_t)row * TLEN;

  // Stage both rows into LDS (float4) while accumulating moments.
  float sp = 0.f, spp = 0.f, st = 0.f, stt = 0.f;
  const float4* p4 = (const float4*)prow;
  const float4* t4 = (const float4*)trow;
  float4* tcv = (float4*)(tcExt + 16);   // +64B keeps 16B alignment
  float4* pcv = (float4*)pcL;
  for (int it = 0; it < TLEN / 128; ++it) {
    int i4 = lane + 32 * it;
    float4 a = p4[i4];
    float4 b = t4[i4];
    sp  += a.x + a.y + a.z + a.w;
    spp += a.x * a.x + a.y * a.y + a.z * a.z + a.w * a.w;
    st  += b.x + b.y + b.z + b.w;
    stt += b.x * b.x + b.y * b.y + b.z * b.z + b.w * b.w;
    pcv[i4] = a;
    tcv[i4] = b;
  }
  if (lane < 16) {
    tcExt[lane]              = trow[TLEN - 16 + lane];  // front halo
    tcExt[16 + TLEN + lane]  = trow[lane];              // back halo
  }
  __syncthreads();

  // Wave-wide butterfly reductions; all lanes end with full sums.
  for (int m = 16; m >= 1; m >>= 1) {
    sp  += __shfl_xor(sp,  m, 32);
    spp += __shfl_xor(spp, m, 32);
    st  += __shfl_xor(st,  m, 32);
    stt += __shfl_xor(stt, m, 32);
  }

  // Per-lane WMMA fragment coordinates:
  //   A: a[p] = A[q][p + 2*hf],  B: b[p] = B[p + 2*hf][q],  D: d[v] = D[v + 8*hf][q]
  const int hf = lane >> 4;
  const int q  = lane & 15;
  const int lb = 32 * hf + q;

  v8f acc0 = {};  // A[i][kk] = tc[16m + i]        -> lag i-j
  v8f accP = {};  // A[i][kk] = tc[16m + i + 16]   -> lag i-j+16 (use i<j)
  v8f accM = {};  // A[i][kk] = tc[16m + i - 16]   -> lag i-j-16 (use i>j)
  for (int t = 0; t < TLEN / 64; ++t) {
    int o = 64 * t + lb;
    float x0 = tcExt[o];        // tc[... - 16]  (halo-offset array: tcExt[x] = tgt[(x-16) mod T])
    float x1 = tcExt[o + 16];
    float x2 = tcExt[o + 32];
    float x3 = tcExt[o + 48];
    float b0 = pcL[o];
    float b1 = pcL[o + 16];
    v2f a0 = {x1, x2};
    v2f ap = {x2, x3};
    v2f am = {x0, x1};
    v2f bb = {b0, b1};
    acc0 = __builtin_amdgcn_wmma_f32_16x16x4_f32(false, a0, false, bb, (short)0, acc0, false, false);
    accP = __builtin_amdgcn_wmma_f32_16x16x4_f32(false, ap, false, bb, (short)0, accP, false, false);
    accM = __builtin_amdgcn_wmma_f32_16x16x4_f32(false, am, false, bb, (short)0, accM, false, false);
  }

  // Spill tiles to LDS for cross-lane diagonal extraction.
#pragma unroll
  for (int v = 0; v < 8; ++v) {
    int i = v + 8 * hf;
    G0s[i * 16 + q] = acc0[v];
    GPs[i * 16 + q] = accP[v];
    GMs[i * 16 + q] = accM[v];
  }
  __syncthreads();

  // Lane s (0..30) sums lag (s-15): 16-|lag| entries from G0 diagonal plus
  // |lag| wrap entries from GP (lag>0) or GM (lag<0). Fixed order -> deterministic.
  float r = -2.0f;
  if (lane < 31) {
    int lag = lane - MAXSHIFT;
    float xc = 0.f;
    if (lag >= 0) {
      for (int n = 0; n + lag <= 15; ++n) xc += G0s[(n + lag) * 16 + n];
      for (int i = 0; i < lag; ++i)       xc += GPs[i * 16 + (i + 16 - lag)];
    } else {
      int L = -lag;
      for (int j = L; j <= 15; ++j) xc += G0s[(j - L) * 16 + j];
      for (int j = 0; j < L; ++j)   xc += GMs[(j + 16 - L) * 16 + j];
    }
    const float invT = 1.0f / (float)TLEN;
    float cov  = xc  - sp * st * invT;   // de-mean correction (circular sum)
    float varp = spp - sp * sp * invT;
    float vart = stt - st * st * invT;
    r = cov / (sqrtf(varp + EPSV) * sqrtf(vart + EPSV));
  }
  // Wave max over lanes (lane 31 carries -2, below any Pearson value).
  for (int m = 16; m >= 1; m >>= 1) {
    float o = __shfl_xor(r, m, 32);
    r = fmaxf(r, o);
  }
  if (lane == 0) best_out[row] = fmaxf(r, -1.0f);
}

// Deterministic mean over B best-values, writes scalar loss.
__global__ __launch_bounds__(256) void finalize_kernel(
    const float* __restrict__ best, float* __restrict__ out, int B) {
  __shared__ float sm[256];
  int tid = threadIdx.x;
  float s = 0.f;
  for (int i = tid; i < B; i += 256) s += best[i];
  sm[tid] = s;
  __syncthreads();
  for (int k = 128; k > 0; k >>= 1) {
    if (tid < k) sm[tid] += sm[tid + k];
    __syncthreads();
  }
  if (tid == 0) out[0] = 1.0f - sm[0] / (float)B;
}

extern "C" void kernel_launch(void* const* d_in, const int* in_sizes, int n_in,
                              void* d_out, int out_size, void* d_ws, size_t ws_size,
                              hipStream_t stream) {
  const float* preds   = (const float*)d_in[0];
  const float* targets = (const float*)d_in[1];
  float* out = (float*)d_out;
  float* ws  = (float*)d_ws;   // B floats: per-row best Pearson

  const int B = in_sizes[0] / TLEN;

  pearson_rows_kernel<<<B, 32, 0, stream>>>(preds, targets, ws);
  finalize_kernel<<<1, 256, 0, stream>>>(ws, out, B);
}